// ScaledDotProductAttention_5248450036165
// MI455X (gfx1250) — compile-verified
//
#include <hip/hip_runtime.h>
#include <math.h>

#define B_ 16
#define S_ 2048
#define D_ 64
#define MT 16                 // query rows per workgroup
#define SSTR 2068             // padded LDS row stride in floats (16B-aligned rows, conflict-free)
#define INV_TEMP 0.125f       // 1 / sqrt(64)

typedef float v2f __attribute__((ext_vector_type(2)));
typedef float v4f __attribute__((ext_vector_type(4)));
typedef float v8f __attribute__((ext_vector_type(8)));

__global__ __launch_bounds__(256)
void attn_wmma_f32_kernel(const float* __restrict__ q,
                          const float* __restrict__ k,
                          const float* __restrict__ v,
                          const int*   __restrict__ mask,
                          float* __restrict__ out_o,
                          float* __restrict__ out_p)
{
    extern __shared__ float smem[];
    float* sS   = smem;                    // [MT][SSTR] masked scores -> probabilities
    float* oBuf = smem + MT * SSTR;        // [2][16][64] K-split partial O

    const int blk   = blockIdx.x;
    const int b     = blk >> 7;            // / (S_/MT)
    const int qbase = (blk & 127) * MT;

    const int tid  = threadIdx.x;
    const int w    = tid >> 5;             // wave id 0..7
    const int lane = tid & 31;
    const int lrow = lane & 15;            // M (A/C) or N (B/C) index
    const int koff = (lane >> 4) << 1;     // K sub-offset: 0 or 2
    const int mhalf = (lane >> 4) << 3;    // C/D row half: 0 or 8

    // ---------------- Q fragments in WMMA-A layout, pre-scaled ----------------
    v2f qa[16];
    {
        const float* qp = q + ((size_t)(b * S_) + qbase + lrow) * D_;
#pragma unroll
        for (int c = 0; c < 16; ++c) {
            v2f t = *(const v2f*)(qp + 4 * c + koff);
            qa[c].x = t.x * INV_TEMP;
            qa[c].y = t.y * INV_TEMP;
        }
    }

    // ---------------- Phase 1: S = (Q/T) K^T, mask, stage in LDS --------------
    for (int t = w; t < S_ / 16; t += 8) {
        const int nbase = t * 16;
        const float* kp = k + ((size_t)(b * S_) + nbase + lrow) * D_;
        __builtin_prefetch(kp + 8 * 16 * D_, 0, 0);   // next tile for this wave

        v8f acc = {0.f, 0.f, 0.f, 0.f, 0.f, 0.f, 0.f, 0.f};
#pragma unroll
        for (int c = 0; c < 16; ++c) {
            v2f kb = *(const v2f*)(kp + 4 * c + koff);  // B[k][n] = K[n][k]
            acc = __builtin_amdgcn_wmma_f32_16x16x4_f32(
                false, qa[c], false, kb, (short)0, acc, false, false);
        }
        const int kcol = nbase + lrow;
#pragma unroll
        for (int j = 0; j < 8; ++j) {
            const int ml = j + mhalf;
            const int mv = mask[(size_t)(qbase + ml) * S_ + kcol];
            sS[ml * SSTR + kcol] = mv ? acc[j] : -1e9f;
        }
    }
    __syncthreads();

    // ---------------- Phase 2: row softmax from LDS, stream attn to HBM -------
    {
        const int r   = tid >> 4;      // row 0..15 (16 threads per row)
        const int sub = tid & 15;
        float* row = sS + r * SSTR;

        float m = -3.4e38f;
#pragma unroll 4
        for (int i = 0; i < 32; ++i) {
            v4f x = *(v4f*)(row + sub * 4 + i * 64);
            m = fmaxf(m, fmaxf(fmaxf(x.x, x.y), fmaxf(x.z, x.w)));
        }
#pragma unroll
        for (int off = 8; off; off >>= 1)
            m = fmaxf(m, __shfl_xor(m, off, 16));

        float ssum = 0.f;
#pragma unroll 4
        for (int i = 0; i < 32; ++i) {
            v4f x = *(v4f*)(row + sub * 4 + i * 64);
            v4f e;
            e.x = __expf(x.x - m); e.y = __expf(x.y - m);
            e.z = __expf(x.z - m); e.w = __expf(x.w - m);
            *(v4f*)(row + sub * 4 + i * 64) = e;
            ssum += (e.x + e.y) + (e.z + e.w);
        }
#pragma unroll
        for (int off = 8; off; off >>= 1)
            ssum += __shfl_xor(ssum, off, 16);
        const float inv = 1.0f / ssum;

        float* pout = out_p + ((size_t)(b * S_) + qbase + r) * S_;
#pragma unroll 2
        for (int i = 0; i < 32; ++i) {
            v4f x = *(v4f*)(row + sub * 4 + i * 64);
            x.x *= inv; x.y *= inv; x.z *= inv; x.w *= inv;
            *(v4f*)(row + sub * 4 + i * 64) = x;        // keep P for phase 3
            *(v4f*)(pout + sub * 4 + i * 64) = x;       // bandwidth-critical store
        }
    }
    __syncthreads();

    // ---------------- Phase 3: O = P V (WMMA f32, 4 N-tiles x 2 K-halves) -----
    {
        const int ntile  = w & 3;
        const int khalf  = w >> 2;
        const int nb     = ntile * 16;
        const int kstart = khalf * (S_ / 2);
        const float* vp  = v + (size_t)(b * S_) * D_ + nb + lrow;  // column base

        v8f acc = {0.f, 0.f, 0.f, 0.f, 0.f, 0.f, 0.f, 0.f};
        for (int kc = 0; kc < (S_ / 2) / 4; ++kc) {
            const int kb4 = kstart + kc * 4;
            v2f a = *(const v2f*)(sS + lrow * SSTR + kb4 + koff);  // A = P chunk
            v2f bb;                                                // B[k][n] = V[k][n]
            bb.x = vp[(size_t)(kb4 + koff) * D_];
            bb.y = vp[(size_t)(kb4 + koff + 1) * D_];
            acc = __builtin_amdgcn_wmma_f32_16x16x4_f32(
                false, a, false, bb, (short)0, acc, false, false);
        }
#pragma unroll
        for (int j = 0; j < 8; ++j)
            oBuf[(khalf * 16 + j + mhalf) * 64 + nb + lrow] = acc[j];
    }
    __syncthreads();

    // ---------------- Combine K-halves, write O -------------------------------
    {
        const int idx = tid * 4;                 // 256 threads x 4 = 16*64
        v4f a0 = *(v4f*)(oBuf + idx);
        v4f a1 = *(v4f*)(oBuf + 1024 + idx);
        v4f o;
        o.x = a0.x + a1.x; o.y = a0.y + a1.y;
        o.z = a0.z + a1.z; o.w = a0.w + a1.w;
        const int rrow = idx >> 6;
        const int col  = idx & 63;
        *(v4f*)(out_o + ((size_t)(b * S_) + qbase + rrow) * D_ + col) = o;
    }
}

extern "C" void kernel_launch(void* const* d_in, const int* in_sizes, int n_in,
                              void* d_out, int out_size, void* d_ws, size_t ws_size,
                              hipStream_t stream) {
    (void)in_sizes; (void)n_in; (void)out_size; (void)d_ws; (void)ws_size;
    const float* q    = (const float*)d_in[0];
    const float* k    = (const float*)d_in[1];
    const float* v    = (const float*)d_in[2];
    const int*   mask = (const int*)d_in[3];

    float* out_o = (float*)d_out;                          // [B,S,D]
    float* out_p = out_o + (size_t)B_ * S_ * D_;           // [B,S,S]

    const size_t shmem = (size_t)(MT * SSTR + 2 * 16 * 64) * sizeof(float);
    dim3 grid(B_ * (S_ / MT));   // 2048 workgroups
    attn_wmma_f32_kernel<<<grid, 256, shmem, stream>>>(q, k, v, mask, out_o, out_p);
}